// TemporalGAT_34600256537209
// MI455X (gfx1250) — compile-verified
//
#include <hip/hip_runtime.h>
#include <hip/hip_bf16.h>

// ---------------- problem constants (match reference) ----------------
#define NNODES 50000
#define NEDGES 800000
#define FDIM   128      // F_IN == HID == H*D
#define NHEAD  4
#define DHEAD  32
#define NEG_SLOPE 0.2f

typedef __attribute__((ext_vector_type(2))) float v2f;
typedef __attribute__((ext_vector_type(8))) float v8f;

// ---------------- small helpers ----------------
__device__ __forceinline__ unsigned f2ord(float f) {
  unsigned u = __float_as_uint(f);
  return (u & 0x80000000u) ? ~u : (u | 0x80000000u);
}
__device__ __forceinline__ float ord2f(unsigned o) {
  unsigned u = (o & 0x80000000u) ? (o ^ 0x80000000u) : ~o;
  return __uint_as_float(u);
}
#define ORD_NEG_INF 0x007FFFFFu   // f2ord(-inf)

__global__ void fill_u32_kernel(unsigned* __restrict__ p, unsigned v, int n) {
  int i = blockIdx.x * blockDim.x + threadIdx.x;
  if (i < n) p[i] = v;
}

// ---------------- WMMA GEMM: C[M,128] = A[M,128] * B[128,128] ----------------
// One block = 16-row stripe; 8 waves, each wave owns one 16x16 output tile.
// f32 WMMA 16x16x4 keeps full precision; workload is HBM-bound so the lower
// per-instruction K is free. A stripe staged into LDS via the CDNA5 async
// global->LDS DMA path (ASYNCcnt); B (weights, 64KB) hits L2/WGP$.
__global__ void __launch_bounds__(256)
gemm128_wmma_f32(const float* __restrict__ A, const float* __restrict__ B,
                 float* __restrict__ C) {
  __shared__ float sA[16 * FDIM];   // 8 KB
  const int tid  = threadIdx.x;
  const int row0 = blockIdx.x * 16;

  // stage 16x128 f32 stripe: 256 threads x 16B x 2 = 8192B
  const float* gsrc = A + (size_t)row0 * FDIM + tid * 4;
#if defined(__HIP_DEVICE_COMPILE__) && defined(__gfx1250__)
  // flat shared-aperture pointer: low 32 bits are the LDS byte offset
  unsigned ldsd = (unsigned)(uintptr_t)(&sA[0]) + (unsigned)tid * 16u;
  // INST_OFFSET is added to BOTH the LDS and the global address (ISA 10.7),
  // so one extra issue at offset:4096 covers the second half of the stripe.
  asm volatile("global_load_async_to_lds_b128 %0, %1, off"
               :: "v"(ldsd), "v"(gsrc) : "memory");
  asm volatile("global_load_async_to_lds_b128 %0, %1, off offset:4096"
               :: "v"(ldsd), "v"(gsrc) : "memory");
  asm volatile("s_wait_asynccnt 0x0" ::: "memory");
#else
  float4* sA4 = (float4*)sA;
  sA4[tid]       = ((const float4*)gsrc)[0];
  sA4[tid + 256] = *(const float4*)(gsrc + 1024);
#endif
  __syncthreads();

  const int lane = tid & 31;
  const int wave = tid >> 5;
  const int col0 = wave * 16;
  const int n    = lane & 15;          // A row (M) / B column within tile
  const int kg   = (lane >> 4) * 2;    // K sub-offset per half-wave

  v8f acc = {};
#pragma unroll
  for (int k = 0; k < FDIM; k += 4) {
    v2f a, b;
    const float2 av = *(const float2*)&sA[n * FDIM + k + kg];   // ds_load_b64
    a.x = av.x; a.y = av.y;
    b.x = B[(k + kg)     * FDIM + col0 + n];
    b.y = B[(k + kg + 1) * FDIM + col0 + n];
    acc = __builtin_amdgcn_wmma_f32_16x16x4_f32(
        /*neg_a=*/false, a, /*neg_b=*/false, b,
        /*c_mod=*/(short)0, acc, /*reuse_a=*/false, /*reuse_b=*/false);
  }

  const int mg = (lane >> 4) * 8;      // C layout: VGPR v -> M = v + 8*(lane>>4)
#pragma unroll
  for (int v = 0; v < 8; ++v)
    C[(size_t)(row0 + mg + v) * FDIM + col0 + n] = acc[v];
}

// ---------------- attention logits per node/head ----------------
__global__ void alpha_kernel(const float* __restrict__ h,
                             const float* __restrict__ a_src,
                             const float* __restrict__ a_dst,
                             float* __restrict__ as, float* __restrict__ ad) {
  int i = blockIdx.x * blockDim.x + threadIdx.x;   // over N*H
  if (i >= NNODES * NHEAD) return;
  int n_ = i >> 2, hh = i & 3;
  const float* hp = h + (size_t)n_ * FDIM + hh * DHEAD;
  const float* s  = a_src + hh * DHEAD;
  const float* d  = a_dst + hh * DHEAD;
  float accs = 0.f, accd = 0.f;
#pragma unroll
  for (int j = 0; j < DHEAD; ++j) { float v = hp[j]; accs += v * s[j]; accd += v * d[j]; }
  as[i] = accs; ad[i] = accd;
}

__device__ __forceinline__ float edge_logit(const float* as, const float* ad,
                                            int s_, int d_, int hh) {
  float a = as[s_ * NHEAD + hh] + ad[d_ * NHEAD + hh];
  return a > 0.f ? a : NEG_SLOPE * a;
}

// ---------------- segment softmax pass 1: max ----------------
__global__ void edge_max_kernel(const int* __restrict__ src, const int* __restrict__ dst,
                                const float* __restrict__ as, const float* __restrict__ ad,
                                unsigned* __restrict__ mOrd) {
  int i = blockIdx.x * blockDim.x + threadIdx.x;   // over E*H
  if (i >= NEDGES * NHEAD) return;
  int e_ = i >> 2, hh = i & 3;
  int s_ = src[e_], d_ = dst[e_];
  atomicMax(&mOrd[d_ * NHEAD + hh], f2ord(edge_logit(as, ad, s_, d_, hh)));
}

// ---------------- pass 2: exp-sum ----------------
__global__ void edge_expsum_kernel(const int* __restrict__ src, const int* __restrict__ dst,
                                   const float* __restrict__ as, const float* __restrict__ ad,
                                   const unsigned* __restrict__ mOrd,
                                   float* __restrict__ ssum) {
  int i = blockIdx.x * blockDim.x + threadIdx.x;
  if (i >= NEDGES * NHEAD) return;
  int e_ = i >> 2, hh = i & 3;
  int s_ = src[e_], d_ = dst[e_];
  float lg = edge_logit(as, ad, s_, d_, hh);
  float m  = ord2f(mOrd[d_ * NHEAD + hh]);
  atomicAdd(&ssum[d_ * NHEAD + hh], __expf(lg - m));
}

// ---------------- pass 3: weighted scatter-aggregate ----------------
// one wave32 per edge; lane owns 4 contiguous features (head = lane>>3)
__global__ void __launch_bounds__(256)
edge_agg_kernel(const int* __restrict__ src, const int* __restrict__ dst,
                const float* __restrict__ as, const float* __restrict__ ad,
                const unsigned* __restrict__ mOrd, const float* __restrict__ ssum,
                const float* __restrict__ h, float* __restrict__ agg) {
  int wid  = (blockIdx.x * blockDim.x + threadIdx.x) >> 5;
  int lane = threadIdx.x & 31;
  if (wid >= NEDGES) return;
  int s_ = src[wid], d_ = dst[wid];
  int hh = lane >> 3;
  float lg = edge_logit(as, ad, s_, d_, hh);
  float m  = ord2f(mOrd[d_ * NHEAD + hh]);
  float alpha = __expf(lg - m) / (ssum[d_ * NHEAD + hh] + 1e-16f);
  const float4 hv = *(const float4*)(h + (size_t)s_ * FDIM + lane * 4);
  float* op = agg + (size_t)d_ * FDIM + lane * 4;
  atomicAdd(op + 0, hv.x * alpha);
  atomicAdd(op + 1, hv.y * alpha);
  atomicAdd(op + 2, hv.z * alpha);
  atomicAdd(op + 3, hv.w * alpha);
}

// ---------------- batch-norm statistics (per channel) ----------------
// blockDim.x == C (32 or 128); each thread owns a channel, strides over rows.
__global__ void stats_kernel(const float* __restrict__ x, const float* __restrict__ bias,
                             int C, int rows, float* __restrict__ sums /*[2C]*/) {
  int c = threadIdx.x;
  float b = bias ? bias[c] : 0.f;
  float s = 0.f, sq = 0.f;
  for (int r = blockIdx.x; r < rows; r += gridDim.x) {
    float v = x[(size_t)r * C + c] + b;
    s += v; sq += v * v;
  }
  atomicAdd(&sums[c], s);
  atomicAdd(&sums[C + c], sq);
}

__global__ void finalize_stats_kernel(float* __restrict__ sums, int C, float inv_n) {
  int c = threadIdx.x;
  if (c >= C) return;
  float mu  = sums[c] * inv_n;
  float var = sums[C + c] * inv_n - mu * mu;
  sums[c]     = mu;
  sums[C + c] = rsqrtf(var + 1e-5f);
}

// ---------------- bn + elu apply (layer 1, in place, C=128) ----------------
__global__ void bn_elu_kernel(float* __restrict__ x, const float* __restrict__ bias,
                              const float* __restrict__ sums,
                              const float* __restrict__ gamma, const float* __restrict__ beta,
                              int total) {
  int i = blockIdx.x * blockDim.x + threadIdx.x;
  if (i >= total) return;
  int c = i & (FDIM - 1);
  float v = x[i] + bias[c];
  v = (v - sums[c]) * sums[FDIM + c] * gamma[c] + beta[c];
  x[i] = v > 0.f ? v : (__expf(v) - 1.f);
}

// ---------------- head mean + bias2 (layer 2, concat=False) ----------------
__global__ void head_mean_kernel(const float* __restrict__ agg,
                                 const float* __restrict__ bias2,
                                 float* __restrict__ out32) {
  int i = blockIdx.x * blockDim.x + threadIdx.x;   // over N*32
  if (i >= NNODES * DHEAD) return;
  int n_ = i >> 5, d = i & 31;
  const float* p = agg + (size_t)n_ * FDIM;
  out32[i] = 0.25f * (p[d] + p[32 + d] + p[64 + d] + p[96 + d]) + bias2[d];
}

// ---------------- final: bn + elu + temporal gating (wave per node) -------
__global__ void __launch_bounds__(256)
final_kernel(const float* __restrict__ x32, const float* __restrict__ sums,
             const float* __restrict__ g, const float* __restrict__ b,
             const float* __restrict__ node_time,
             const float* __restrict__ tw1, const float* __restrict__ tb1,
             const float* __restrict__ tw2, const float* __restrict__ tb2,
             float* __restrict__ out) {
  int gid = blockIdx.x * blockDim.x + threadIdx.x;
  int n_  = gid >> 5;
  int d   = gid & 31;                 // D == wave32 width
  if (n_ >= NNODES) return;
  float v = x32[(size_t)n_ * DHEAD + d];
  v = (v - sums[d]) * sums[DHEAD + d] * g[d] + b[d];
  v = v > 0.f ? v : (__expf(v) - 1.f);
  // temporal gate: t = relu(time*w1+b1) [32]; z = t.w2 + b2; sigmoid
  float t = node_time[n_] * tw1[d] + tb1[d];
  t = t > 0.f ? t : 0.f;
  float z = t * tw2[d];
#pragma unroll
  for (int off = 16; off > 0; off >>= 1) z += __shfl_xor(z, off, 32);
  float gate = 1.f / (1.f + __expf(-(z + tb2[0])));
  out[(size_t)n_ * DHEAD + d] = v * gate;
}

// =======================================================================
extern "C" void kernel_launch(void* const* d_in, const int* in_sizes, int n_in,
                              void* d_out, int out_size, void* d_ws, size_t ws_size,
                              hipStream_t stream) {
  const float* x        = (const float*)d_in[0];
  const int*   eidx     = (const int*)  d_in[1];
  const float* etime    = (const float*)d_in[2];
  const float* W1       = (const float*)d_in[3];
  const float* att_src1 = (const float*)d_in[4];
  const float* att_dst1 = (const float*)d_in[5];
  const float* bias1    = (const float*)d_in[6];
  const float* W2       = (const float*)d_in[7];
  const float* att_src2 = (const float*)d_in[8];
  const float* att_dst2 = (const float*)d_in[9];
  const float* bias2    = (const float*)d_in[10];
  const float* bn1_g    = (const float*)d_in[11];
  const float* bn1_b    = (const float*)d_in[12];
  const float* bn2_g    = (const float*)d_in[13];
  const float* bn2_b    = (const float*)d_in[14];
  const float* t_w1     = (const float*)d_in[15];
  const float* t_b1     = (const float*)d_in[16];
  const float* t_w2     = (const float*)d_in[17];
  const float* t_b2     = (const float*)d_in[18];
  const int* src = eidx;
  const int* dst = eidx + NEDGES;
  float* out = (float*)d_out;

  // ---- workspace layout (~61 MB) ----
  float*    bufH   = (float*)d_ws;                         // [N,128] GEMM out
  float*    bufAgg = bufH   + (size_t)NNODES * FDIM;       // [N,128] agg / X2
  float*    as     = bufAgg + (size_t)NNODES * FDIM;       // [N,4]
  float*    ad     = as     + (size_t)NNODES * NHEAD;      // [N,4]
  unsigned* mOrd   = (unsigned*)(ad + (size_t)NNODES * NHEAD); // [N,4]
  float*    ssum   = (float*)(mOrd + (size_t)NNODES * NHEAD);  // [N,4]
  float*    stats  = ssum   + (size_t)NNODES * NHEAD;      // [256]
  float*    buf32  = stats  + 256;                         // [N,32]

  const int NH  = NNODES * NHEAD;
  const int EH  = NEDGES * NHEAD;
  const int NC  = NNODES * FDIM;
  const int ND  = NNODES * DHEAD;
  auto blk = [](int n) { return (n + 255) / 256; };

  // =========================== LAYER 1 ===========================
  fill_u32_kernel<<<blk(NC), 256, 0, stream>>>((unsigned*)bufAgg, 0u, NC);
  fill_u32_kernel<<<blk(NH), 256, 0, stream>>>(mOrd, ORD_NEG_INF, NH);
  fill_u32_kernel<<<blk(NH), 256, 0, stream>>>((unsigned*)ssum, 0u, NH);
  fill_u32_kernel<<<1, 256, 0, stream>>>((unsigned*)stats, 0u, 256);

  gemm128_wmma_f32<<<NNODES / 16, 256, 0, stream>>>(x, W1, bufH);
  alpha_kernel<<<blk(NH), 256, 0, stream>>>(bufH, att_src1, att_dst1, as, ad);
  edge_max_kernel<<<blk(EH), 256, 0, stream>>>(src, dst, as, ad, mOrd);
  edge_expsum_kernel<<<blk(EH), 256, 0, stream>>>(src, dst, as, ad, mOrd, ssum);
  edge_agg_kernel<<<NEDGES * 32 / 256, 256, 0, stream>>>(src, dst, as, ad, mOrd, ssum,
                                                         bufH, bufAgg);
  stats_kernel<<<256, FDIM, 0, stream>>>(bufAgg, bias1, FDIM, NNODES, stats);
  finalize_stats_kernel<<<1, FDIM, 0, stream>>>(stats, FDIM, 1.0f / NNODES);
  bn_elu_kernel<<<blk(NC), 256, 0, stream>>>(bufAgg, bias1, stats, bn1_g, bn1_b, NC);

  // =========================== LAYER 2 ===========================
  gemm128_wmma_f32<<<NNODES / 16, 256, 0, stream>>>(bufAgg, W2, bufH);
  // bufAgg consumed by GEMM2 (stream-ordered) -> reuse as agg2
  fill_u32_kernel<<<blk(NC), 256, 0, stream>>>((unsigned*)bufAgg, 0u, NC);
  fill_u32_kernel<<<blk(NH), 256, 0, stream>>>(mOrd, ORD_NEG_INF, NH);
  fill_u32_kernel<<<blk(NH), 256, 0, stream>>>((unsigned*)ssum, 0u, NH);
  fill_u32_kernel<<<1, 256, 0, stream>>>((unsigned*)stats, 0u, 256);

  alpha_kernel<<<blk(NH), 256, 0, stream>>>(bufH, att_src2, att_dst2, as, ad);
  edge_max_kernel<<<blk(EH), 256, 0, stream>>>(src, dst, as, ad, mOrd);
  edge_expsum_kernel<<<blk(EH), 256, 0, stream>>>(src, dst, as, ad, mOrd, ssum);
  edge_agg_kernel<<<NEDGES * 32 / 256, 256, 0, stream>>>(src, dst, as, ad, mOrd, ssum,
                                                         bufH, bufAgg);
  head_mean_kernel<<<blk(ND), 256, 0, stream>>>(bufAgg, bias2, buf32);
  stats_kernel<<<256, DHEAD, 0, stream>>>(buf32, nullptr, DHEAD, NNODES, stats);
  finalize_stats_kernel<<<1, DHEAD, 0, stream>>>(stats, DHEAD, 1.0f / NNODES);
  final_kernel<<<blk(ND), 256, 0, stream>>>(buf32, stats, bn2_g, bn2_b, etime,
                                            t_w1, t_b1, t_w2, t_b2, out);
}